// Generator_73229192397059
// MI455X (gfx1250) — compile-verified
//
#include <hip/hip_runtime.h>
#include <hip/hip_bf16.h>

// ---------------------------------------------------------------------------
// Problem constants (match the reference harness)
// ---------------------------------------------------------------------------
#define NPH    64
#define PLEN   320
#define HDIM   1280
#define BATCH  16
#define WSAMP  480000

typedef __attribute__((ext_vector_type(2))) float v2f;
typedef __attribute__((ext_vector_type(8))) float v8f;

// ---------------------------------------------------------------------------
// WMMA f32 GEMM:  out[64,N] = act( A[64,K] @ W[K,N] + bias[N] )
//   act = 0 -> relu, 1 -> tanh
// Block: 128 threads = 4 wave32s. Wave w owns rows [16w, 16w+16).
// Grid.x: N/16 column tiles; the K x 16 weight slab for this column tile is
// staged into dynamic LDS (up to 80KB, CDNA5 WGP has 320KB).
// Uses V_WMMA_F32_16X16X4_F32 (native fp32 matrix pipe, exact f32 semantics).
// ---------------------------------------------------------------------------
__global__ void mlp_gemm_wmma(const float* __restrict__ A,
                              const float* __restrict__ Wt,
                              const float* __restrict__ bias,
                              float* __restrict__ out,
                              int K, int N, int act) {
    extern __shared__ float Bs[];                 // [K x 16] weight slab
    const int tid  = threadIdx.x;
    const int lane = tid & 31;
    const int wave = tid >> 5;                    // 0..3 -> M tile
    const int n0   = blockIdx.x * 16;

    // Cooperative stage of the K x 16 column block of W into LDS.
    // K*16 is a multiple of 128 for all layers -> no tail divergence.
    for (int idx = tid; idx < K * 16; idx += 128) {
        int k = idx >> 4;
        int n = idx & 15;
        Bs[idx] = Wt[k * N + n0 + n];
    }
    __syncthreads();

    const int half = (lane < 16) ? 0 : 2;         // ISA 16x4 f32 A layout:
    const int l16  = lane & 15;                   // lanes 0-15: K{0,1}, 16-31: K{2,3}

    // C/D init: bias depends only on the N column -> same value in all 8 regs.
    float bv = bias[n0 + l16];
    v8f c = { bv, bv, bv, bv, bv, bv, bv, bv };

    const float* Arow = A + (wave * 16 + l16) * K;

    for (int k = 0; k < K; k += 4) {
        v2f a, b;
        a.x = Arow[k + half + 0];
        a.y = Arow[k + half + 1];
        b.x = Bs[(k + half + 0) * 16 + l16];
        b.y = Bs[(k + half + 1) * 16 + l16];
        // 8 args: (neg_a, A, neg_b, B, c_mod, C, reuse_a, reuse_b)
        c = __builtin_amdgcn_wmma_f32_16x16x4_f32(
                false, a, false, b, (short)0, c, false, false);
    }

    // D layout: VGPR r holds row (r) for lanes 0-15 and row (r+8) for 16-31.
#pragma unroll
    for (int r = 0; r < 8; ++r) {
        int m   = wave * 16 + r + ((lane < 16) ? 0 : 8);
        float v = c[r];
        v = act ? tanhf(v) : fmaxf(v, 0.0f);
        out[m * N + n0 + l16] = v;
    }
}

// ---------------------------------------------------------------------------
// Zero d_out (16 x 480000 f32 = 30.7MB) with 128-bit stores.
// ---------------------------------------------------------------------------
__global__ void zero_out_kernel(float4* __restrict__ out, int n4) {
    int i      = blockIdx.x * blockDim.x + threadIdx.x;
    int stride = gridDim.x * blockDim.x;
    float4 z = make_float4(0.0f, 0.0f, 0.0f, 0.0f);
    for (; i < n4; i += stride) out[i] = z;
}

// ---------------------------------------------------------------------------
// Tile scatter: one 320-thread block (10 wave32s) per tile record.
//   win = 0.5*(1 - cos(2*pi*(woff+j)/wlen));  atomicAdd into d_out directly
// (C == 1, so the broadcast output IS the accumulator; batch==BATCH rows are
//  the padded dummy sink and are skipped).
// ---------------------------------------------------------------------------
__global__ void tile_scatter_kernel(const float* __restrict__ pdict,
                                    const int* __restrict__ tb,
                                    const int* __restrict__ ts,
                                    const int* __restrict__ tp,
                                    const int* __restrict__ tw,
                                    const int* __restrict__ tl,
                                    float* __restrict__ out) {
    const int t = blockIdx.x;
    const int b = tb[t];
    if (b >= BATCH) return;                       // dummy sink row

    const int   s    = ts[t];
    const int   phn  = tp[t];
    const float woff = (float)tw[t];
    const float wlen = (float)tl[t];
    const int   j    = threadIdx.x;

    const float pos = woff + (float)j;
    const float win = 0.5f * (1.0f - __cosf(6.28318530717958647692f * pos / wlen));
    const float val = pdict[phn * PLEN + j] * win;

    atomicAdd(out + (size_t)b * WSAMP + (size_t)(s + j), val);
}

// ---------------------------------------------------------------------------
// Launch: 4 GEMM layers ping-ponging activations through d_ws, then zero the
// output and scatter all tiles into it.
//   d_in order: data, phonemes, w1,b1, w2,b2, w3,b3, w4,b4,
//               tile_batch, tile_start, tile_phn, tile_woff, tile_wlen
// ---------------------------------------------------------------------------
extern "C" void kernel_launch(void* const* d_in, const int* in_sizes, int n_in,
                              void* d_out, int out_size, void* d_ws, size_t ws_size,
                              hipStream_t stream) {
    const float* phonemes = (const float*)d_in[1];
    const float* w1 = (const float*)d_in[2];
    const float* b1 = (const float*)d_in[3];
    const float* w2 = (const float*)d_in[4];
    const float* b2 = (const float*)d_in[5];
    const float* w3 = (const float*)d_in[6];
    const float* b3 = (const float*)d_in[7];
    const float* w4 = (const float*)d_in[8];
    const float* b4 = (const float*)d_in[9];
    const int* tb = (const int*)d_in[10];
    const int* ts = (const int*)d_in[11];
    const int* tp = (const int*)d_in[12];
    const int* tw = (const int*)d_in[13];
    const int* tl = (const int*)d_in[14];
    const int  T  = in_sizes[10];                 // T_MAX tile records

    float* out = (float*)d_out;

    // Workspace layout: x1, x2, x3 (64x1280 each), pdict (64x320)
    float* x1    = (float*)d_ws;
    float* x2    = x1 + NPH * HDIM;
    float* x3    = x2 + NPH * HDIM;
    float* pdict = x3 + NPH * HDIM;

    // Layer 1: [64,320] @ [320,1280], relu.   LDS slab: 320*16*4  = 20KB
    mlp_gemm_wmma<<<HDIM / 16, 128, PLEN * 16 * sizeof(float), stream>>>(
        phonemes, w1, b1, x1, PLEN, HDIM, 0);
    // Layer 2: [64,1280] @ [1280,1280], relu. LDS slab: 1280*16*4 = 80KB
    mlp_gemm_wmma<<<HDIM / 16, 128, HDIM * 16 * sizeof(float), stream>>>(
        x1, w2, b2, x2, HDIM, HDIM, 0);
    // Layer 3: [64,1280] @ [1280,1280], relu
    mlp_gemm_wmma<<<HDIM / 16, 128, HDIM * 16 * sizeof(float), stream>>>(
        x2, w3, b3, x3, HDIM, HDIM, 0);
    // Layer 4: [64,1280] @ [1280,320], tanh -> pdict
    mlp_gemm_wmma<<<PLEN / 16, 128, HDIM * 16 * sizeof(float), stream>>>(
        x3, w4, b4, pdict, HDIM, PLEN, 1);

    // Zero the accumulator/output (16*480000 floats, divisible by 4).
    zero_out_kernel<<<1920, 256, 0, stream>>>((float4*)out,
                                              BATCH * WSAMP / 4);

    // Scatter all tile records with f32 global atomics.
    tile_scatter_kernel<<<T, PLEN, 0, stream>>>(pdict, tb, ts, tp, tw, tl, out);
}